// EncModule_23845658427827
// MI455X (gfx1250) — compile-verified
//
#include <hip/hip_runtime.h>
#include <hip/hip_bf16.h>

#define EPS 1e-5f

typedef __attribute__((ext_vector_type(2))) float v2f;
typedef __attribute__((ext_vector_type(8))) float v8f;

static __device__ __forceinline__ v8f wmma_f32(v2f a, v2f b, v8f c) {
  // D = A(16x4) * B(4x16) + C(16x16), all f32, wave32
  return __builtin_amdgcn_wmma_f32_16x16x4_f32(false, a, false, b, (short)0, c, false, false);
}

// async global -> LDS copy, 16 bytes per lane.  lds_dst must point into __shared__
// memory: the ptrtoint both yields the LDS byte offset (low 32 bits of the flat
// shared-aperture address) and captures the object so the compiler must assume
// the asm writes it.  Tracked by ASYNCcnt.
static __device__ __forceinline__ void async_copy_b128(const float* lds_dst, const float* gp) {
  unsigned ldsoff = (unsigned)(uintptr_t)lds_dst;
  unsigned long long ga = (unsigned long long)(uintptr_t)gp;
  asm volatile("global_load_async_to_lds_b128 %0, %1, off"
               :: "v"(ldsoff), "v"(ga) : "memory");
}
static __device__ __forceinline__ void wait_async_le1() {
  asm volatile("s_wait_asynccnt 0x1" ::: "memory");
}
static __device__ __forceinline__ void wait_async_0() {
  asm volatile("s_wait_asynccnt 0x0" ::: "memory");
}

// ---------------- constants ----------------
#define BB 8
#define CC 512
#define NN 16384
#define KK 32
#define NCHUNK 32               // N split for aggregation partials
#define CHUNKLEN (NN / NCHUNK)  // 512
#define XPAD 72                 // LDS row pitch (floats): 2*72 % 64 == 16 -> halves on disjoint banks

// ---------------- kernel 0: init (c2 + zero asum) ----------------
__global__ void k_init(const float* __restrict__ codewords, float* __restrict__ c2,
                       float* __restrict__ asum) {
  int t = threadIdx.x;
  if (t < BB * KK) asum[t] = 0.f;
  if (t < KK) {
    float s = 0.f;
    const float* row = codewords + (size_t)t * CC;
    for (int c = 0; c < CC; ++c) { float v = row[c]; s += v * v; }
    c2[t] = s;
  }
}

// ---------------- kernel 1: 1x1 conv GEMM + BN2 + ReLU -> feat [B][C][N] ----------------
// Block = 8 waves: o-range 128 (8 x 16-row bands), n-strip 64 wide.
// X tile (16c x 64n) staged in LDS via async copies, double-buffered, shared by all waves.
__global__ __launch_bounds__(256) void k_conv(
    const float* __restrict__ x, const float* __restrict__ W,
    const float* __restrict__ g2, const float* __restrict__ b2,
    const float* __restrict__ m2, const float* __restrict__ v2,
    float* __restrict__ feat) {
  __shared__ float xs[2][16 * XPAD];

  int blk  = blockIdx.x;            // 8192 blocks
  int b    = blk >> 10;             // /1024
  int rem  = blk & 1023;
  int ob   = rem >> 8;              // 0..3  (128-row o band)
  int nt   = rem & 255;             // 0..255 (64-col n strip)
  int tid  = threadIdx.x;
  int wave = tid >> 5;
  int lane = tid & 31;
  int l16  = lane & 15;
  int hi   = lane >> 4;             // 0/1 half
  int o0   = ob * 128 + wave * 16;
  int n0   = nt * 64;

  const float* Xb   = x + (size_t)b * CC * NN;
  const float* Wrow = W + (size_t)(o0 + l16) * CC + hi * 2;

  // cooperative copy mapping: thread -> (row, 4-float column chunk)
  int crow = tid >> 4;              // 0..15
  int ccol = (tid & 15) * 4;        // 0..60
  const float* gsrc = Xb + (size_t)crow * NN + n0 + ccol;   // + cc0*NN per step
  float* ldst0 = &xs[0][crow * XPAD + ccol];
  float* ldst1 = &xs[1][crow * XPAD + ccol];

  v8f acc0 = {}, acc1 = {}, acc2 = {}, acc3 = {};

  const int STEPS = CC / 16;        // 32
  // prime buffer 0
  async_copy_b128(ldst0, gsrc);
  for (int s = 0; s < STEPS; ++s) {
    __syncthreads();                              // everyone done computing on buf[(s+1)&1]
    if (s + 1 < STEPS) {
      async_copy_b128(((s + 1) & 1) ? ldst1 : ldst0, gsrc + (size_t)(s + 1) * 16 * NN);
      wait_async_le1();                           // copy(s) complete (in-order)
    } else {
      wait_async_0();
    }
    __syncthreads();                              // copy(s) visible to all waves
    const float* xb = &xs[s & 1][hi * 2 * XPAD + l16];
    int cc0 = s * 16;
#pragma unroll
    for (int cp = 0; cp < 16; cp += 4) {
      v2f a = *(const v2f*)(Wrow + cc0 + cp);     // W[o][c..c+1] (contiguous pair)
      const float* xr = xb + cp * XPAD;
      v2f b0, b1, b2v, b3;
      b0.x  = xr[0];  b0.y  = xr[XPAD];
      b1.x  = xr[16]; b1.y  = xr[XPAD + 16];
      b2v.x = xr[32]; b2v.y = xr[XPAD + 32];
      b3.x  = xr[48]; b3.y  = xr[XPAD + 48];
      acc0 = wmma_f32(a, b0, acc0);
      acc1 = wmma_f32(a, b1, acc1);
      acc2 = wmma_f32(a, b2v, acc2);
      acc3 = wmma_f32(a, b3, acc3);
    }
  }
  // BN (eval) + ReLU + store
#pragma unroll
  for (int r = 0; r < 8; ++r) {
    int o = o0 + r + hi * 8;
    float sc = rsqrtf(v2[o] + EPS) * g2[o];
    float mu = m2[o];
    float bt = b2[o];
    float* dst = feat + ((size_t)b * CC + o) * NN + n0 + l16;
    dst[0]  = fmaxf((acc0[r] - mu) * sc + bt, 0.f);
    dst[16] = fmaxf((acc1[r] - mu) * sc + bt, 0.f);
    dst[32] = fmaxf((acc2[r] - mu) * sc + bt, 0.f);
    dst[48] = fmaxf((acc3[r] - mu) * sc + bt, 0.f);
  }
}

// ---------------- kernel 2a: scaled-L2 softmax assignment -> aw_t [B][K][N], asum [B][K] ----------------
__global__ __launch_bounds__(256) void k_assign(
    const float* __restrict__ feat, const float* __restrict__ codewords,
    const float* __restrict__ scale, const float* __restrict__ c2,
    float* __restrict__ aw_t, float* __restrict__ asum) {
  int blk  = blockIdx.x;           // 1024 blocks
  int b    = blk >> 7;
  int rem  = blk & 127;
  int wave = threadIdx.x >> 5;
  int lane = threadIdx.x & 31;
  int l16  = lane & 15;
  int hi   = lane >> 4;
  int n0   = (rem * 8 + wave) * 16;

  const float* Fb = feat + (size_t)b * CC * NN;
  v8f acc0 = {}, acc1 = {};
  float s2 = 0.f;
  for (int c = 0; c < CC; c += 4) {
    const float* Fr = Fb + (size_t)(c + hi * 2) * NN + n0 + l16;
    v2f a; a.x = Fr[0]; a.y = Fr[NN];
    s2 += a.x * a.x + a.y * a.y;
    const float* cw0 = codewords + (size_t)l16 * CC + c + hi * 2;
    const float* cw1 = codewords + (size_t)(16 + l16) * CC + c + hi * 2;
    v2f b0 = *(const v2f*)cw0;
    v2f b1 = *(const v2f*)cw1;
    acc0 = wmma_f32(a, b0, acc0);   // xc for k = 0..15
    acc1 = wmma_f32(a, b1, acc1);   // xc for k = 16..31
  }
  // x2 per row: each lane holds partial over its 2 c-columns; pair lanes l and l+16
  float x2all = s2 + __shfl_xor(s2, 16, 32);       // = x2[n0 + (lane&15)], dup in both halves
  float sc0 = scale[l16], sc1 = scale[16 + l16];
  float c20 = c2[l16],    c21 = c2[16 + l16];

  float awsum0 = 0.f, awsum1 = 0.f;
  size_t awbase = (size_t)b * KK * NN;
#pragma unroll
  for (int r = 0; r < 8; ++r) {
    int nrow = n0 + r + hi * 8;
    float x2r = __shfl(x2all, r + hi * 8, 32);     // broadcast x2 of this row to its half
    float d0 = sc0 * (x2r - 2.f * acc0[r] + c20);  // k = l16
    float d1 = sc1 * (x2r - 2.f * acc1[r] + c21);  // k = 16+l16
    // softmax over 32 k values: 16 lanes of this half x 2 per lane
    float m = fmaxf(d0, d1);
    m = fmaxf(m, __shfl_xor(m, 1, 32));
    m = fmaxf(m, __shfl_xor(m, 2, 32));
    m = fmaxf(m, __shfl_xor(m, 4, 32));
    m = fmaxf(m, __shfl_xor(m, 8, 32));
    float e0 = __expf(d0 - m), e1 = __expf(d1 - m);
    float s = e0 + e1;
    s += __shfl_xor(s, 1, 32);
    s += __shfl_xor(s, 2, 32);
    s += __shfl_xor(s, 4, 32);
    s += __shfl_xor(s, 8, 32);
    float inv = 1.f / s;
    float a0 = e0 * inv, a1 = e1 * inv;
    awsum0 += a0; awsum1 += a1;
    aw_t[awbase + (size_t)l16 * NN + nrow]        = a0;
    aw_t[awbase + (size_t)(16 + l16) * NN + nrow] = a1;
  }
  awsum0 += __shfl_xor(awsum0, 16, 32);
  awsum1 += __shfl_xor(awsum1, 16, 32);
  if (lane < 16) {
    atomicAdd(&asum[b * KK + l16], awsum0);
    atomicAdd(&asum[b * KK + 16 + l16], awsum1);
  }
}

// ---------------- kernel 2b: aggregation GEMM partials: encpart[b][chunk][k][c] ----------------
__global__ __launch_bounds__(256) void k_aggregate(
    const float* __restrict__ aw_t, const float* __restrict__ feat,
    float* __restrict__ encpart) {
  int blk   = blockIdx.x;          // 512 blocks: b(8) x chunk(32) x chalf(2)
  int b     = blk >> 6;
  int rem   = blk & 63;
  int chunk = rem >> 1;
  int chalf = rem & 1;
  int wave  = threadIdx.x >> 5;
  int lane  = threadIdx.x & 31;
  int l16   = lane & 15;
  int hi    = lane >> 4;
  int kband = wave >> 2;           // 0/1
  int cg    = wave & 3;
  int k0    = kband * 16;
  int c0    = chalf * 256 + cg * 64;
  int nstart = chunk * CHUNKLEN;

  const float* awrow = aw_t + ((size_t)b * KK + k0 + l16) * NN;
  const float* f0 = feat + ((size_t)b * CC + c0 + 0  + l16) * NN;
  const float* f1 = feat + ((size_t)b * CC + c0 + 16 + l16) * NN;
  const float* f2 = feat + ((size_t)b * CC + c0 + 32 + l16) * NN;
  const float* f3 = feat + ((size_t)b * CC + c0 + 48 + l16) * NN;

  v8f acc0 = {}, acc1 = {}, acc2 = {}, acc3 = {};
  for (int n = nstart; n < nstart + CHUNKLEN; n += 4) {
    int np = n + hi * 2;
    v2f a  = *(const v2f*)(awrow + np);      // aw_t[k][n], aw_t[k][n+1]
    v2f b0 = *(const v2f*)(f0 + np);         // feat[c][n], feat[c][n+1]
    v2f b1 = *(const v2f*)(f1 + np);
    v2f b2v = *(const v2f*)(f2 + np);
    v2f b3 = *(const v2f*)(f3 + np);
    acc0 = wmma_f32(a, b0, acc0);
    acc1 = wmma_f32(a, b1, acc1);
    acc2 = wmma_f32(a, b2v, acc2);
    acc3 = wmma_f32(a, b3, acc3);
  }
#pragma unroll
  for (int r = 0; r < 8; ++r) {
    int k = k0 + r + hi * 8;
    float* dst = encpart + (((size_t)(b * NCHUNK + chunk) * KK + k) * CC + c0 + l16);
    dst[0]  = acc0[r];
    dst[16] = acc1[r];
    dst[32] = acc2[r];
    dst[48] = acc3[r];
  }
}

// ---------------- kernel 3: reduce chunks, -asum*codewords, BN1+ReLU, mean over K ----------------
__global__ __launch_bounds__(512) void k_reduce(
    const float* __restrict__ encpart, const float* __restrict__ asum,
    const float* __restrict__ codewords,
    const float* __restrict__ g1, const float* __restrict__ b1,
    const float* __restrict__ m1, const float* __restrict__ v1,
    float* __restrict__ ef_out, float* __restrict__ ef_ws) {
  int b = blockIdx.x;
  int c = threadIdx.x;
  float accum = 0.f;
  for (int k = 0; k < KK; ++k) {
    float v = 0.f;
    const float* p = encpart + (((size_t)(b * NCHUNK) * KK + k) * CC + c);
    for (int ch = 0; ch < NCHUNK; ++ch) v += p[(size_t)ch * KK * CC];
    v -= asum[b * KK + k] * codewords[(size_t)k * CC + c];
    float bn = (v - m1[k]) * (rsqrtf(v1[k] + EPS) * g1[k]) + b1[k];
    accum += fmaxf(bn, 0.f);
  }
  float ef = accum * (1.f / KK);
  ef_out[b * CC + c] = ef;
  ef_ws[b * CC + c] = ef;
}

// ---------------- kernel 4: fc + sigmoid -> gamma [B][C] ----------------
__global__ __launch_bounds__(256) void k_gate(
    const float* __restrict__ ef, const float* __restrict__ fc_w,
    const float* __restrict__ fc_b, float* __restrict__ gamma) {
  int t = blockIdx.x * 256 + threadIdx.x;   // 0 .. 4095
  int b = t >> 9;
  int o = t & 511;
  const float* e = ef + b * CC;
  const float* w = fc_w + (size_t)o * CC;
  float dot = 0.f;
  for (int c = 0; c < CC; ++c) dot += e[c] * w[c];
  float z = dot + fc_b[o];
  gamma[t] = 1.f / (1.f + __expf(-z));
}

// ---------------- kernel 5: residual gate: out = relu(x * (1 + gamma)) ----------------
__global__ __launch_bounds__(256) void k_residual(
    const float* __restrict__ x, const float* __restrict__ gamma,
    float* __restrict__ out) {
  size_t i4 = (size_t)blockIdx.x * 256 + threadIdx.x;   // float4 index
  float g = 1.f + gamma[(int)(i4 >> 12)];               // (b*C + c) = i4 / (N/4)
  const float4 xv = ((const float4*)x)[i4];
  float4 o;
  o.x = fmaxf(xv.x * g, 0.f);
  o.y = fmaxf(xv.y * g, 0.f);
  o.z = fmaxf(xv.z * g, 0.f);
  o.w = fmaxf(xv.w * g, 0.f);
  ((float4*)out)[i4] = o;
}

extern "C" void kernel_launch(void* const* d_in, const int* in_sizes, int n_in,
                              void* d_out, int out_size, void* d_ws, size_t ws_size,
                              hipStream_t stream) {
  const float* x        = (const float*)d_in[0];
  const float* conv_w   = (const float*)d_in[1];
  const float* bn2_g    = (const float*)d_in[2];
  const float* bn2_b    = (const float*)d_in[3];
  const float* bn2_m    = (const float*)d_in[4];
  const float* bn2_v    = (const float*)d_in[5];
  const float* codew    = (const float*)d_in[6];
  const float* scale    = (const float*)d_in[7];
  const float* bn1_g    = (const float*)d_in[8];
  const float* bn1_b    = (const float*)d_in[9];
  const float* bn1_m    = (const float*)d_in[10];
  const float* bn1_v    = (const float*)d_in[11];
  const float* fc_w     = (const float*)d_in[12];
  const float* fc_b     = (const float*)d_in[13];

  float* out_ef  = (float*)d_out;                 // [B,C] = 4096 floats
  float* out_img = out_ef + BB * CC;              // [B,C,H,W]

  // workspace layout (floats)
  float* ws = (float*)d_ws;
  float* feat    = ws;                                     // 67,108,864
  float* aw_t    = feat + (size_t)BB * CC * NN;            // 4,194,304
  float* encpart = aw_t + (size_t)BB * KK * NN;            // 4,194,304
  float* asum    = encpart + (size_t)BB * NCHUNK * KK * CC;// 256
  float* c2      = asum + BB * KK;                         // 32
  float* ef_ws   = c2 + KK;                                // 4096
  float* gamma   = ef_ws + BB * CC;                        // 4096

  k_init<<<1, 256, 0, stream>>>(codew, c2, asum);
  k_conv<<<8192, 256, 0, stream>>>(x, conv_w, bn2_g, bn2_b, bn2_m, bn2_v, feat);
  k_assign<<<1024, 256, 0, stream>>>(feat, codew, scale, c2, aw_t, asum);
  k_aggregate<<<512, 256, 0, stream>>>(aw_t, feat, encpart);
  k_reduce<<<BB, 512, 0, stream>>>(encpart, asum, codew, bn1_g, bn1_b, bn1_m, bn1_v,
                                   out_ef, ef_ws);
  k_gate<<<16, 256, 0, stream>>>(ef_ws, fc_w, fc_b, gamma);
  k_residual<<<(BB * CC * NN / 4) / 256, 256, 0, stream>>>(x, gamma, out_img);
}